// QuadrotorDynamics_43508018708685
// MI455X (gfx1250) — compile-verified
//
#include <hip/hip_runtime.h>
#include <hip/hip_bf16.h>
#include <math.h>

// Only check on the device pass: amdgcn builtins aren't declared when
// compiling the host (x86) side of the HIP TU.
#if defined(__HIP_DEVICE_COMPILE__)
#if !__has_builtin(__builtin_amdgcn_wmma_f32_16x16x4_f32)
#error "missing __builtin_amdgcn_wmma_f32_16x16x4_f32 on this toolchain"
#endif
#endif

typedef __attribute__((ext_vector_type(2))) float v2f;
typedef __attribute__((ext_vector_type(8))) float v8f;

#define T_LEN 80     // LB + PS
#define TT_LEN 28
#define WPAD 96      // 16-col zero prefix + 80 data
#define W4 48        // 16-col zero prefix + 28 data + 4 zero tail

struct NetParams {
  const float* w[4];
  const float* b[4];
  const float* g[4];
  const float* be[4];
  const float* mn[4];
  const float* vr[4];
  const float* fc1_w;
  const float* fc1_b;
  const float* fc2_w;
  const float* fc2_b;
  int od;       // 4 (motor) or 6 (accel)
  int res_off;  // slot in shared result buffer
};

// Dilated causal conv block + bias + BN + ReLU, as 16x16-tiled GEMM using
// V_WMMA_F32_16X16X4_F32. Reduction dim q = ci*8 (k fastest), chunks of 4.
// A (weights): lane(16h+p): M=p, VGPR v -> K = v + 2h.
// B (activations from LDS, padded layout): N=p, VGPR v -> K = v + 2h.
// D: row = r + 8h, col = p.
__device__ __forceinline__ void conv_block(
    const float* __restrict__ wgt, const float* __restrict__ bias,
    const float* __restrict__ g, const float* __restrict__ be,
    const float* __restrict__ mn, const float* __restrict__ vr,
    const float* src, int srcW,          // LDS src, data starts at col 16
    float* dst, int dstW, int dstOff,    // LDS dst: dst[m*dstW + dstOff + t]
    int ci, int co, int Lout, int p, int h)
{
  const int KC = ci * 2;  // (ci*8)/4 chunks of K=4
  for (int cb = 0; cb < co; cb += 16) {
    for (int t0 = 0; t0 < Lout; t0 += 16) {
      v8f acc = {0.f, 0.f, 0.f, 0.f, 0.f, 0.f, 0.f, 0.f};
      for (int kc = 0; kc < KC; ++kc) {
        int q0 = kc * 4 + 2 * h;   // K index for VGPR 0 of this lane-half
        int q1 = q0 + 1;           // K index for VGPR 1
        v2f a, bb;
        a.x = wgt[((cb + p) * ci + (q0 >> 3)) * 8 + (q0 & 7)];
        a.y = wgt[((cb + p) * ci + (q1 >> 3)) * 8 + (q1 & 7)];
        // xpad[ci, t + 2k] lives at LDS col (t + 2k + 2) (14 causal zeros)
        bb.x = src[(q0 >> 3) * srcW + t0 + p + 2 * (q0 & 7) + 2];
        bb.y = src[(q1 >> 3) * srcW + t0 + p + 2 * (q1 & 7) + 2];
        acc = __builtin_amdgcn_wmma_f32_16x16x4_f32(
            false, a, false, bb, (short)0, acc, false, false);
      }
      // epilogue: bias + batchnorm + relu, store to LDS
      int t = t0 + p;
      for (int r = 0; r < 8; ++r) {
        int m = cb + r + 8 * h;
        float sc  = g[m] * rsqrtf(vr[m] + 1e-5f);
        float val = (acc[r] + bias[m] - mn[m]) * sc + be[m];
        val = val > 0.f ? val : 0.f;
        if (t < Lout) dst[m * dstW + dstOff + t] = val;
      }
    }
  }
}

// One full TCN executed by a single wave32. bufA: 16*96 floats, bufB: 32*96.
__device__ void run_net(const NetParams& P, const float* __restrict__ xg,
                        float* bufA, float* bufB, float* res, int lane)
{
  const int p = lane & 15, h = lane >> 4;

  for (int i = lane; i < 16 * WPAD; i += 32) bufA[i] = 0.f;
  for (int i = lane; i < 32 * WPAD; i += 32) bufB[i] = 0.f;
  // load x (16 x 80) into padded layout
  for (int i = lane; i < 16 * T_LEN; i += 32) {
    int c = i / T_LEN, t = i - c * T_LEN;
    bufA[c * WPAD + 16 + t] = xg[i];
  }
  __builtin_amdgcn_wave_barrier();

  // block 1: 16 -> 16, bufA -> bufB
  conv_block(P.w[0], P.b[0], P.g[0], P.be[0], P.mn[0], P.vr[0],
             bufA, WPAD, bufB, WPAD, 16, 16, 16, T_LEN, p, h);
  __builtin_amdgcn_wave_barrier();
  // block 2: 16 -> 16, bufB -> bufA
  conv_block(P.w[1], P.b[1], P.g[1], P.be[1], P.mn[1], P.vr[1],
             bufB, WPAD, bufA, WPAD, 16, 16, 16, T_LEN, p, h);
  __builtin_amdgcn_wave_barrier();
  // block 3: 16 -> 32, bufA -> bufB
  conv_block(P.w[2], P.b[2], P.g[2], P.be[2], P.mn[2], P.vr[2],
             bufA, WPAD, bufB, WPAD, 16, 16, 32, T_LEN, p, h);
  __builtin_amdgcn_wave_barrier();

  // block 4 input: last TT cols of h3, repacked 32 x W4 into bufA (fits: 32*48==16*96)
  for (int i = lane; i < 32 * W4; i += 32) bufA[i] = 0.f;
  __builtin_amdgcn_wave_barrier();
  for (int i = lane; i < 32 * TT_LEN; i += 32) {
    int c = i / TT_LEN, t = i - c * TT_LEN;
    bufA[c * W4 + 16 + t] = bufB[c * WPAD + 16 + (T_LEN - TT_LEN) + t];
  }
  __builtin_amdgcn_wave_barrier();
  // block 4: 32 -> 16, bufA -> bufB as flat h4[c*28 + t] (matches reshape order)
  conv_block(P.w[3], P.b[3], P.g[3], P.be[3], P.mn[3], P.vr[3],
             bufA, W4, bufB, TT_LEN, 0, 32, 16, TT_LEN, p, h);
  __builtin_amdgcn_wave_barrier();

  // FC1: 448 -> 128 matvec (N=1 -> plain f32 dots), relu, result in bufA[0..127]
  for (int j = 0; j < 4; ++j) {
    int o = lane + 32 * j;
    float acc = P.fc1_b[o];
    const float* wr = P.fc1_w + o * 448;
    for (int i = 0; i < 448; ++i) acc += wr[i] * bufB[i];
    bufA[o] = acc > 0.f ? acc : 0.f;
  }
  __builtin_amdgcn_wave_barrier();

  // FC2: 128 -> od
  if (lane < P.od) {
    float acc = P.fc2_b[lane];
    const float* wr = P.fc2_w + lane * 128;
    for (int i = 0; i < 128; ++i) acc += wr[i] * bufA[i];
    res[P.res_off + lane] = acc;
  }
}

__global__ void __launch_bounds__(64)
quad_kernel(NetParams PM, NetParams PA,
            const float* __restrict__ xg, float* __restrict__ out)
{
  __shared__ float sA[2][16 * WPAD];
  __shared__ float sB[2][32 * WPAD];
  __shared__ float res[16];

  const int tid = threadIdx.x;
  const int wave = tid >> 5;
  const int lane = tid & 31;

  if (wave == 0) run_net(PM, xg, sA[0], sB[0], res, lane);
  else           run_net(PA, xg, sA[1], sB[1], res, lane);
  __syncthreads();

  // zero output everywhere except column T-2 (written only by thread 0)
  for (int i = tid; i < 16 * T_LEN; i += 64)
    if ((i % T_LEN) != (T_LEN - 2)) out[i] = 0.f;

  if (tid == 0) {
    const float L_ARM = 0.2f, MASS = 0.8f, DCOEF = 0.02f;
    const float KT = 0.1f, KR = 0.1f;
    const float IXX = 0.005f, IYY = 0.005f, IZZ = 0.01f;

    float a0 = xg[0 * T_LEN + 78], a1 = xg[1 * T_LEN + 78], a2 = xg[2 * T_LEN + 78];
    float r0 = xg[6 * T_LEN + 78], r1 = xg[7 * T_LEN + 78], r2 = xg[8 * T_LEN + 78];
    float v0 = xg[9 * T_LEN + 78], v1 = xg[10 * T_LEN + 78], v2 = xg[11 * T_LEN + 78];

    float th0 = res[0], th1 = res[1], th2 = res[2], th3 = res[3];
    float cL = 0.707f * L_ARM;
    float tq0 = th0 + th1 + th2 + th3;
    float tq1 = cL * (th0 - th1 - th2 + th3);
    float tq2 = cL * (-th0 - th1 + th2 + th3);
    float tq3 = DCOEF * (-th0 + th1 - th2 + th3);

    float sp = sinf(a0), cp = cosf(a0);
    float st = sinf(a1), ct = cosf(a1);
    float ss = sinf(a2), cs = cosf(a2);

    // vel_dot = rbi[:,2] * (tq0/MASS) - KT*vel - g
    float f = tq0 / MASS;
    float vd0 = (cp * cs * st + sp * ss) * f - KT * v0;
    float vd1 = (cp * ss * st - sp * cs) * f - KT * v1;
    float vd2 = (ct * cp) * f - KT * v2 - 9.8067f;

    // ang_dot = M^-1 rate  (closed form, det of 2x2 block = ct)
    float ad1 = cp * r1 - sp * r2;
    float ad2 = (sp * r1 + cp * r2) / ct;
    float ad0 = r0 + sp * ad2;

    // rate_dot = I^-1 (torques[1:] - rate x (I rate) - KR*rate)
    float cx = r1 * r2 * (IZZ - IYY);
    float cy = r0 * r2 * (IXX - IZZ);
    float cz = r0 * r1 * (IYY - IXX);
    float rd0 = (tq1 - cx - KR * r0) / IXX;
    float rd1 = (tq2 - cy - KR * r1) / IYY;
    float rd2 = (tq3 - cz - KR * r2) / IZZ;

    // add accel-net output to rows 6..11
    rd0 += res[8];  rd1 += res[9];  rd2 += res[10];
    vd0 += res[11]; vd1 += res[12]; vd2 += res[13];

    float sd[16] = {ad0, ad1, ad2, v0, v1, v2,
                    rd0, rd1, rd2, vd0, vd1, vd2,
                    0.f, 0.f, 0.f, 0.f};
    for (int c = 0; c < 16; ++c) out[c * T_LEN + (T_LEN - 2)] = sd[c];
  }
}

extern "C" void kernel_launch(void* const* d_in, const int* in_sizes, int n_in,
                              void* d_out, int out_size, void* d_ws, size_t ws_size,
                              hipStream_t stream) {
  (void)in_sizes; (void)out_size; (void)d_ws; (void)ws_size;
  const float* xg = (const float*)d_in[1];  // d_in[0] = t (unused by reference)

  NetParams PM, PA;
  // Sorted-leaf order (jax tree flatten of dict):
  // b1-4, be1-4, fc1_b, fc1_w, fc2_b, fc2_w, g1-4, m1-4, v1-4, w1-4
  auto fill_leaves = [](NetParams& P, void* const* L) {
    for (int i = 0; i < 4; ++i) P.b[i]  = (const float*)L[0 + i];
    for (int i = 0; i < 4; ++i) P.be[i] = (const float*)L[4 + i];
    P.fc1_b = (const float*)L[8];
    P.fc1_w = (const float*)L[9];
    P.fc2_b = (const float*)L[10];
    P.fc2_w = (const float*)L[11];
    for (int i = 0; i < 4; ++i) P.g[i]  = (const float*)L[12 + i];
    for (int i = 0; i < 4; ++i) P.mn[i] = (const float*)L[16 + i];
    for (int i = 0; i < 4; ++i) P.vr[i] = (const float*)L[20 + i];
    for (int i = 0; i < 4; ++i) P.w[i]  = (const float*)L[24 + i];
  };
  auto fill_flat = [](NetParams& P, int od, const float* base) {
    const int cos_[4] = {16, 16, 32, 16};
    const int wsz[4]  = {2048, 2048, 4096, 4096};
    size_t off = 0;
    for (int i = 0; i < 4; ++i) { P.b[i]  = base + off; off += cos_[i]; }
    for (int i = 0; i < 4; ++i) { P.be[i] = base + off; off += cos_[i]; }
    P.fc1_b = base + off; off += 128;
    P.fc1_w = base + off; off += 57344;
    P.fc2_b = base + off; off += od;
    P.fc2_w = base + off; off += (size_t)od * 128;
    for (int i = 0; i < 4; ++i) { P.g[i]  = base + off; off += cos_[i]; }
    for (int i = 0; i < 4; ++i) { P.mn[i] = base + off; off += cos_[i]; }
    for (int i = 0; i < 4; ++i) { P.vr[i] = base + off; off += cos_[i]; }
    for (int i = 0; i < 4; ++i) { P.w[i]  = base + off; off += wsz[i]; }
  };

  if (n_in >= 58) {            // each dict leaf is a separate input
    fill_leaves(PM, d_in + 2);
    fill_leaves(PA, d_in + 2 + 28);
  } else {                     // params dicts passed as flat concatenations
    fill_flat(PM, 4, (const float*)d_in[2]);
    fill_flat(PA, 6, (const float*)d_in[3]);
  }
  PM.od = 4; PM.res_off = 0;
  PA.od = 6; PA.res_off = 8;

  quad_kernel<<<1, 64, 0, stream>>>(PM, PA, xg, (float*)d_out);
}